// BatteryRNNCell_82523501626123
// MI455X (gfx1250) — compile-verified
//
#include <hip/hip_runtime.h>
#include <hip/hip_bf16.h>

typedef float f32x4 __attribute__((ext_vector_type(4)));
typedef float f32x2 __attribute__((ext_vector_type(2)));

#define DEVFN static __device__ __forceinline__

DEVFN float fast_rcp(float x)  { return __builtin_amdgcn_rcpf(x); }
DEVFN float fast_sqrt(float x) { return __builtin_amdgcn_sqrtf(x); }
DEVFN float fast_rsq(float x)  { return __builtin_amdgcn_rsqf(x); }
DEVFN float fast_log2(float x) { return __builtin_amdgcn_logf(x); }   // v_log_f32 (log2)

// Cross the two 16-lane halves of the wave32 (lane <- lane^16).
// Preferred: v_permlanex16_b32 with identity selects (pure VALU, no DS wait).
// Fallback: ds_swizzle SWAPX16 (xor=0x10,and=0x1f -> offset 0x401f).
DEVFN float xhalf(float v) {
#if __has_builtin(__builtin_amdgcn_permlanex16)
    int x = __builtin_bit_cast(int, v);
    int r = __builtin_amdgcn_permlanex16(x, x, 0x76543210, 0xFEDCBA98, false, false);
    return __builtin_bit_cast(float, r);
#else
    return __builtin_bit_cast(float,
        __builtin_amdgcn_ds_swizzle(__builtin_bit_cast(int, v), 0x401f));
#endif
}

// Two lanes per battery: lanes 0-15 = positive electrode of batteries base..base+15,
// lanes 16-31 = negative electrode of the same batteries. Both electrodes execute the
// SAME instruction stream (no EXEC divergence) with per-lane constants; the per-step
// voltage is combined with one cross-half permute. 8192 batteries -> 512 wave32s
// (vs 256 one-thread-per-battery), each with ~half the serial per-step work.
__global__ __launch_bounds__(32)
void battery_scan_split(const float* __restrict__ current,
                        const float* __restrict__ init_state,
                        const float* __restrict__ Ap_scale,
                        const float* __restrict__ An0_scale,
                        float* __restrict__ out,
                        int B, int T)
{
    const int  lane  = (int)threadIdx.x;        // 0..31 (wave32)
    const bool half0 = (lane < 16);             // true = positive electrode
    const int  bl    = lane & 15;
    int b = blockIdx.x * 16 + bl;
    const bool bOK = (b < B);
    if (!bOK) b = B - 1;                        // clamp: keep all 32 lanes convergent

    // ----- shared constants -----
    const float invF     = 1.0f / 96487.0f;          // 1/FARADAY
    const float invQS    = 1.0f / 1266.6666667f;     // 1/Q_S_MAX
    const float invVOLB  = 1.0f / 1.8e-5f;
    const float invVOLS  = 1.0f / 2.0e-6f;
    const float invTDIFF = 1.0f / 7.0e6f;
    const float invTO    = 1.0f / 6.08671f;
    const float LN2      = 0.6931471805599453f;

    // ----- per-lane (per-electrode) constants -----
    // z = (i/S) / (2*K*sqrt(x(1-x)))  ->  i * c1 * rsq(x(1-x))
    const float c1    = half0 ? ((1.0f / 0.00030962f)  * (0.5f / 248898.0f))
                              : ((1.0f / 0.000437545f) * (0.5f / 2120.96f));
    const float invTs = half0 ? (1.0f / 46.4311f) : (1.0f / 1001.38f);
    const unsigned smask = half0 ? 0u : 0x80000000u;   // +-i via sign-bit xor

    // ----- per-lane state (init_state is [B,8]); both halves read their battery -----
    const float* st = init_state + (size_t)b * 8;
    float Tb = st[0];
    float Vo = st[1];                 // tracked identically by both halves
    float Vs = st[half0 ? 3 : 2];     // Vsp / Vsn
    float qB = st[half0 ? 6 : 4];     // qpB / qnB
    float qS = st[half0 ? 7 : 5];     // qpS / qnS

    // Tbdot == 0 -> hoist thermal factors (ln2 folded into the log2 coefficients).
    const float kTln2 = (8.3144621f / (96487.0f * 0.5f)) * Tb * LN2;          // RT/(F*a)*ln2
    const float kT2s  = (half0 ? 1.0f : -1.0f) * (8.3144621f / 96487.0f) * Tb * LN2;

    // ----- Redlich-Kister sum as one degree-13 polynomial in t = 2x-1 -----
    // V_k = (A_k/F)[t^{k+1} - k(1-t^2)/2 * t^{k-1}] => P(t) = sum e_m t^m.
    // Electrode sign and U0 folded into the coefficients:
    //   p-lanes: es = +e, es[0] += U0P ; n-lanes: es = {-U0N, -an0, 0, ...}.
    const float BASE_AP[13] = { -31593.7f, 0.106747f, 24606.4f, -78561.9f,
        13317.9f, 307387.0f, 84916.1f, -1074690.0f, 2285.04f, 990894.0f,
        283920.0f, -161513.0f, -469218.0f };
    float a[13];
    #pragma unroll
    for (int k = 0; k < 13; ++k) a[k] = Ap_scale[k] * BASE_AP[k] * invF; // uniform s_loads
    const float an0 = An0_scale[0] * 86.19f * invF;

    float es[14];
    #pragma unroll
    for (int m = 0; m < 14; ++m) {
        float pv = 0.0f;
        if (m >= 1)  pv += a[m - 1] * (1.0f + 0.5f * (float)(m - 1));
        if (m <= 11) pv -= a[m + 1] * (0.5f * (float)(m + 1));
        if (m == 0)  pv += 4.03f;                                   // U0P
        float nv = (m == 0) ? -0.01f : ((m == 1) ? -an0 : 0.0f);    // -(U0N + an0*t)
        es[m] = half0 ? pv : nv;
    }

    // ----- one Euler step for this lane's electrode; returns this half's val -----
    // (full battery voltage V = val + xhalf(val), combined at chunk level)
    auto step = [&](float i) -> float {
        float x  = qS * invQS;
        float z  = (i * c1) * fast_rsq(x * (1.0f - x));
        float VsNom = kTln2 * fast_log2(z + fast_sqrt(__builtin_fmaf(z, z, 1.0f))); // kT*asinh
        float dBS = (qB * invVOLB - qS * invVOLS) * invTDIFF;
        float si = __builtin_bit_cast(float, __builtin_bit_cast(unsigned, i) ^ smask);
        Vo += (i * 0.117215f - Vo) * invTO;     // identical on both halves
        Vs += (VsNom - Vs) * invTs;
        qB -= dBS;
        qS += dBS + si;                         // qnS += dBS - i ; qpS += dBS + i
        // output from the *next* state
        float x2 = qS * invQS;
        float l2 = fast_log2(fast_rcp(x2) - 1.0f);      // log2((1-x)/x)
        float t  = __builtin_fmaf(2.0f, x2, -1.0f);
        float s = es[13];
        #pragma unroll
        for (int m = 12; m >= 0; --m) s = __builtin_fmaf(s, t, es[m]); // 13-FMA Horner
        float val = __builtin_fmaf(kT2s, l2, s) - Vs;   // sgn*potential - Vs_own
        return __builtin_fmaf(-0.5f, Vo, val);          // split Vo between the halves
    };

    const float* __restrict__ cur = current + (size_t)b * T;
    // Branchless full-EXEC stores: p-half writes t=4c,4c+1 ; n-half writes t=4c+2,4c+3.
    float* __restrict__ obase = out + (size_t)b * T + (half0 ? 0 : 2);

    int t0 = 0;
    if ((T & 3) == 0) {
        const f32x4* __restrict__ c4 = reinterpret_cast<const f32x4*>(cur);
        const int nc = T >> 2;
        for (int c = 0; c < nc; ++c) {
            __builtin_prefetch(c4 + c + 16, 0, 1);   // 256B ahead -> global_prefetch_b8
            f32x4 iv = c4[c];                        // both halves hit the same lines
            f32x4 valv;                              // 4 steps, combines batched after
            valv.x = step(iv.x);
            valv.y = step(iv.y);
            valv.z = step(iv.z);
            valv.w = step(iv.w);
            f32x4 v;                                 // V = val_own + val_partner
            v.x = valv.x + xhalf(valv.x);
            v.y = valv.y + xhalf(valv.y);
            v.z = valv.z + xhalf(valv.z);
            v.w = valv.w + xhalf(valv.w);
            f32x2 sv;                                // both halves hold identical v
            sv.x = half0 ? v.x : v.z;
            sv.y = half0 ? v.y : v.w;
            __builtin_nontemporal_store(sv, reinterpret_cast<f32x2*>(obase + 4 * c));
        }
        t0 = nc << 2;
    }
    for (int t = t0; t < T; ++t) {                   // tail (unused for T=1000)
        float val = step(cur[t]);
        float v = val + xhalf(val);
        if (half0 && bOK)
            __builtin_nontemporal_store(v, out + (size_t)b * T + t);
    }
}

extern "C" void kernel_launch(void* const* d_in, const int* in_sizes, int n_in,
                              void* d_out, int out_size, void* d_ws, size_t ws_size,
                              hipStream_t stream) {
    const float* current    = (const float*)d_in[0];  // [B,T]
    const float* init_state = (const float*)d_in[1];  // [B,8]
    const float* ap_scale   = (const float*)d_in[2];  // [13]
    const float* an0_scale  = (const float*)d_in[3];  // [1]
    float* outp = (float*)d_out;                      // [B,T,1]

    const int B = in_sizes[1] / 8;
    const int T = (B > 0) ? (in_sizes[0] / B) : 0;
    if (B <= 0 || T <= 0) return;

    dim3 block(32);                  // one wave32 per workgroup -> max WGP spread
    dim3 grid((B + 15) / 16);        // 16 batteries per wave (2 lanes per battery)
    battery_scan_split<<<grid, block, 0, stream>>>(current, init_state,
                                                   ap_scale, an0_scale,
                                                   outp, B, T);
}